// CorrelationPytorch_62560493634036
// MI455X (gfx1250) — compile-verified
//
#include <hip/hip_runtime.h>

// Problem constants (from reference)
#define Nb   2
#define Lb   8
#define Cb   128
#define RHW  48
#define HW   2304          // 48*48 (both query HxW and reference RHxRW)
#define DNUM 25            // (2*MAX_DISP+1)^2
#define DLc  (DNUM * Lb)   // 200

typedef __attribute__((ext_vector_type(2))) float v2f;
typedef __attribute__((ext_vector_type(8))) float v8f;

// One wave32 per 16-pixel tile of one (n,l).
// WMMA f32 16x16x4: M = N = 16 pixels of the tile, K = 4 channels/step,
// 32 steps cover C=128 using two independent accumulators (even/odd steps)
// to break the D->C dependence chain. Diagonal of D = per-pixel correlation.
__global__ __launch_bounds__(128) void corr_wmma_kernel(
    const float* __restrict__ input1,     // [N,C,H,W]
    const float* __restrict__ input2,     // [N,L,C,RH,RW]
    const float* __restrict__ qc,         // [N,3,H,W]
    const float* __restrict__ sc,         // [N,L,3,RH,RW]
    const float* __restrict__ P,          // [N,L,3,4]
    float* __restrict__ out)              // corr | coords | mask (flat)
{
    const int lane  = threadIdx.x & 31;
    const int wid   = blockIdx.x * (blockDim.x >> 5) + (threadIdx.x >> 5);
    const int tile  = wid % (HW / 16);
    const int nl    = wid / (HW / 16);
    const int l     = nl % Lb;
    const int n     = nl / Lb;

    const int halfK = lane >> 4;      // 0: K={0,1}, 1: K={2,3} within a step
    const int col   = lane & 15;      // pixel row/col within tile (M and N)
    const int pix   = tile * 16 + col;
    const int cl    = halfK * 2;      // channel sub-offset inside 4-wide K

    // ---- inline projection for this lane's pixel ----
    const float x = qc[(n * 3 + 0) * HW + pix];
    const float y = qc[(n * 3 + 1) * HW + pix];
    const float z = qc[(n * 3 + 2) * HW + pix];
    const float* Pb = P + (n * Lb + l) * 12;
    const float p0 = Pb[0] * x + Pb[1] * y + Pb[2]  * z + Pb[3];
    const float p1 = Pb[4] * x + Pb[5] * y + Pb[6]  * z + Pb[7];
    const float p2 = Pb[8] * x + Pb[9] * y + Pb[10] * z + Pb[11];
    const float zc  = (fabsf(p2) < 1e-8f) ? 1e-8f : p2;
    const float px0 = rintf(p0 / zc);   // RNE == jnp.round
    const float py0 = rintf(p1 / zc);

    // ---- load B (query features) once; reused for all 25 offsets ----
    // B[k][col] = input1[n, 4*s + k, pix]; lane supplies rows cl, cl+1.
    const float* b1 = input1 + (n * Cb) * HW + pix;
    v2f B[32];
#pragma unroll
    for (int s = 0; s < 32; ++s) {
        B[s].x = b1[(4 * s + cl) * HW];
        B[s].y = b1[(4 * s + cl + 1) * HW];
    }

    float* out_corr   = out;
    float* out_coords = out + (size_t)Nb * DLc * HW;        // + 921600
    float* out_mask   = out + (size_t)Nb * DLc * HW * 4;    // + 3686400

    // Lanes 0-7 own diag rows 0-7; lanes 24-31 own diag rows 8-15.
    const bool active = (lane < 8) || (lane >= 24);
    const int  idx    = col & 7;   // vgpr index of this lane's diag element

    for (int o = 0; o < DNUM; ++o) {
        const float dy = (float)(o / 5 - 2);
        const float dx = (float)(o % 5 - 2);
        const float pxf = px0 + dx;
        const float pyf = py0 + dy;
        const float valid = (pxf >= 0.0f && pxf <= (float)(RHW - 1) &&
                             pyf >= 0.0f && pyf <= (float)(RHW - 1)) ? 1.0f : 0.0f;
        const int pxi = (int)fminf(fmaxf(pxf, 0.0f), (float)(RHW - 1));
        const int pyi = (int)fminf(fmaxf(pyf, 0.0f), (float)(RHW - 1));

        // A[m][k] = input2[n, l, 4*s + k, pyi(m), pxi(m)]; this lane is row m=col.
        const float* ap = input2 + ((((n * Lb + l) * Cb + cl) * RHW + pyi) * RHW + pxi);

        // Hoist all 64 gathered loads of this offset -> one deep load clause
        // (maximize memory-level parallelism; data is L2-resident).
        v2f Aa[32];
#pragma unroll
        for (int s = 0; s < 32; ++s) {
            Aa[s].x = ap[s * 4 * HW];        // channel 4s+cl   (immediate offsets)
            Aa[s].y = ap[s * 4 * HW + HW];   // channel 4s+cl+1
        }

        v8f acc0 = {};
        v8f acc1 = {};
#pragma unroll
        for (int s = 0; s < 32; s += 2) {
            acc0 = __builtin_amdgcn_wmma_f32_16x16x4_f32(
                false, Aa[s],     false, B[s],     (short)0, acc0, false, false);
            acc1 = __builtin_amdgcn_wmma_f32_16x16x4_f32(
                false, Aa[s + 1], false, B[s + 1], (short)0, acc1, false, false);
        }

        // diagonal select: element (r,r) sits in vgpr (r&7) of lane r / r+16
        const float c0 = acc0[0] + acc1[0], c1 = acc0[1] + acc1[1];
        const float c2 = acc0[2] + acc1[2], c3 = acc0[3] + acc1[3];
        const float c4 = acc0[4] + acc1[4], c5 = acc0[5] + acc1[5];
        const float c6 = acc0[6] + acc1[6], c7 = acc0[7] + acc1[7];
        const float s0 = (idx & 1) ? c1 : c0;
        const float s1 = (idx & 1) ? c3 : c2;
        const float s2 = (idx & 1) ? c5 : c4;
        const float s3 = (idx & 1) ? c7 : c6;
        const float t0 = (idx & 2) ? s1 : s0;
        const float t1 = (idx & 2) ? s3 : s2;
        const float dg = (idx & 4) ? t1 : t0;

        if (active) {
            const int    ch    = o * Lb + l;
            const size_t cbase = ((size_t)n * DLc + ch) * HW + pix;
            out_corr[cbase] = dg * valid;
            out_mask[cbase] = valid;
            const float* sp = sc + (((n * Lb + l) * 3) * RHW + pyi) * RHW + pxi;
            const size_t obase = (((size_t)n * DLc + ch) * 3) * HW + pix;
            out_coords[obase]          = sp[0]             * valid;
            out_coords[obase + HW]     = sp[RHW * RHW]     * valid;
            out_coords[obase + 2 * HW] = sp[2 * RHW * RHW] * valid;
        }
    }
}

extern "C" void kernel_launch(void* const* d_in, const int* in_sizes, int n_in,
                              void* d_out, int out_size, void* d_ws, size_t ws_size,
                              hipStream_t stream) {
    const float* input1 = (const float*)d_in[0];
    const float* input2 = (const float*)d_in[1];
    const float* qc     = (const float*)d_in[2];
    const float* sc     = (const float*)d_in[3];
    const float* P      = (const float*)d_in[4];
    float* out = (float*)d_out;

    // N*L*(HW/16) = 2*8*144 = 2304 waves; 4 waves (128 threads) per block.
    const int total_waves = Nb * Lb * (HW / 16);
    const int waves_per_block = 4;
    dim3 block(32 * waves_per_block);
    dim3 grid(total_waves / waves_per_block);
    corr_wmma_kernel<<<grid, block, 0, stream>>>(input1, input2, qc, sc, P, out);
}